// Diffusion_layer_LearnableDegOperators_89704686944725
// MI455X (gfx1250) — compile-verified
//
#include <hip/hip_runtime.h>
#include <hip/hip_bf16.h>

#define N_NODES 1024
#define WIDTH   64
#define NB      64          // number of 16-wide panels
#define LEAKY   0.01f

typedef __attribute__((ext_vector_type(2))) float v2f;
typedef __attribute__((ext_vector_type(8))) float v8f;

// D = A(16x4) * B(4x16) + C, f32 WMMA (CDNA5)
static __device__ __forceinline__ v8f wmma4(v2f a, v2f b, v8f c) {
    return __builtin_amdgcn_wmma_f32_16x16x4_f32(
        false, a, false, b, (short)0, c, false, false);
}

// ---------------------------------------------------------------------------
// Op = ga*(Aa Aa^T)/n + gd*(Ad Ad^T)/n + (ga+gd+alpha)*I   (lower triangle)
// One wave per 32x32 output block (2x2 WMMA register tiles: 4 loads -> 4 WMMAs
// per matrix per K-step, 2x the fragment reuse of 1x1 tiling).
// ---------------------------------------------------------------------------
__global__ __launch_bounds__(32)
void k_build_op(const float* __restrict__ Aa, const float* __restrict__ Ad,
                const float* __restrict__ g_adv, const float* __restrict__ g_diff,
                const float* __restrict__ alpha, float* __restrict__ Op) {
    const int bi = blockIdx.x, bj = blockIdx.y;
    if (bj > bi) return;                        // lower triangle only (uniform)
    const int lane = threadIdx.x;
    const int m   = lane & 15;                  // A: row-in-tile, B: col-in-tile
    const int kb2 = (lane >> 4) << 1;           // K sub-offset {0,2}
    const int i0 = bi * 32, j0 = bj * 32;

    const float* ai0 = Aa + (i0 + m) * N_NODES;
    const float* ai1 = ai0 + 16 * N_NODES;
    const float* aj0 = Aa + (j0 + m) * N_NODES;
    const float* aj1 = aj0 + 16 * N_NODES;
    const float* di0 = Ad + (i0 + m) * N_NODES;
    const float* di1 = di0 + 16 * N_NODES;
    const float* dj0 = Ad + (j0 + m) * N_NODES;
    const float* dj1 = dj0 + 16 * N_NODES;

    v8f ca00 = {}, ca01 = {}, ca10 = {}, ca11 = {};
    v8f cd00 = {}, cd01 = {}, cd10 = {}, cd11 = {};
    for (int k0 = 0; k0 < N_NODES; k0 += 4) {
        v2f A0 = *(const v2f*)(ai0 + k0 + kb2);
        v2f A1 = *(const v2f*)(ai1 + k0 + kb2);
        v2f B0 = *(const v2f*)(aj0 + k0 + kb2);
        v2f B1 = *(const v2f*)(aj1 + k0 + kb2);
        ca00 = wmma4(A0, B0, ca00);
        ca01 = wmma4(A0, B1, ca01);
        ca10 = wmma4(A1, B0, ca10);
        ca11 = wmma4(A1, B1, ca11);
        v2f D0 = *(const v2f*)(di0 + k0 + kb2);
        v2f D1 = *(const v2f*)(di1 + k0 + kb2);
        v2f E0 = *(const v2f*)(dj0 + k0 + kb2);
        v2f E1 = *(const v2f*)(dj1 + k0 + kb2);
        cd00 = wmma4(D0, E0, cd00);
        cd01 = wmma4(D0, E1, cd01);
        cd10 = wmma4(D1, E0, cd10);
        cd11 = wmma4(D1, E1, cd11);
    }
    const float ga = g_adv[0], gd = g_diff[0], al = alpha[0];
    const int col = lane & 15;
    const int rb8 = (lane >> 4) * 8;
    const float scale = 1.0f / (float)N_NODES;
    const float diagv = ga + gd + al;

#define STORE_TILE(CA, CD, IOFF, JOFF)                                         \
    for (int r = 0; r < 8; ++r) {                                              \
        int gi = i0 + (IOFF) + r + rb8;                                        \
        int gj = j0 + (JOFF) + col;                                            \
        float v = (ga * (CA)[r] + gd * (CD)[r]) * scale;                       \
        if (gi == gj) v += diagv;                                              \
        Op[gi * N_NODES + gj] = v;                                             \
    }
    STORE_TILE(ca00, cd00, 0, 0)
    STORE_TILE(ca01, cd01, 0, 16)
    STORE_TILE(ca10, cd10, 16, 0)
    STORE_TILE(ca11, cd11, 16, 16)
#undef STORE_TILE
}

// ---------------------------------------------------------------------------
// Fused blocked Cholesky (panel 16), single workgroup of 32 waves.
// Per step j: wave0 factors the diag block wave-synchronously in LDS,
// 1024 threads do the panel TRSM (one row each), then 32 waves sweep the
// trailing rank-16 SYRK update with WMMA (uniform triangular skip).
// ---------------------------------------------------------------------------
__global__ __launch_bounds__(1024)
void k_chol_fused(float* __restrict__ Op) {
    __shared__ float Lb[16][17];
    const int tid  = threadIdx.x;
    const int wave = tid >> 5;
    const int lane = tid & 31;

    for (int j = 0; j < NB; ++j) {
        // --- diagonal block factor (wave 0 only, lockstep within the wave) ---
        if (wave == 0) {
            for (int idx = lane; idx < 256; idx += 32)
                Lb[idx >> 4][idx & 15] =
                    Op[(j * 16 + (idx >> 4)) * N_NODES + j * 16 + (idx & 15)];
            __builtin_amdgcn_wave_barrier();
            for (int kk = 0; kk < 16; ++kk) {
                if (lane == 0) Lb[kk][kk] = sqrtf(fmaxf(Lb[kk][kk], 1e-30f));
                __builtin_amdgcn_wave_barrier();
                if (lane > kk && lane < 16) Lb[lane][kk] /= Lb[kk][kk];
                __builtin_amdgcn_wave_barrier();
                if (lane > kk && lane < 16)
                    for (int c = kk + 1; c <= lane; ++c)
                        Lb[lane][c] -= Lb[lane][kk] * Lb[c][kk];
                __builtin_amdgcn_wave_barrier();
            }
            for (int idx = lane; idx < 256; idx += 32)
                Op[(j * 16 + (idx >> 4)) * N_NODES + j * 16 + (idx & 15)] =
                    Lb[idx >> 4][idx & 15];
        }
        __syncthreads();

        // --- panel TRSM: one row per thread ---
        const int row = 16 * (j + 1) + tid;
        if (row < N_NODES) {
            float x[16];
            float* rp = Op + row * N_NODES + j * 16;
            for (int c = 0; c < 16; ++c) {
                float s = rp[c];
                for (int c2 = 0; c2 < c; ++c2) s -= x[c2] * Lb[c][c2];
                x[c] = s / Lb[c][c];
            }
            for (int c = 0; c < 16; ++c) rp[c] = x[c];
        }
        __threadfence();
        __syncthreads();

        // --- trailing SYRK update, WMMA, tiles distributed over 32 waves ---
        const int nt = NB - 1 - j;
        for (int t = wave; t < nt * nt; t += 32) {
            const int ti = j + 1 + t / nt;
            const int tk = j + 1 + t % nt;
            if (tk > ti) continue;              // uniform per wave
            const int m   = lane & 15;
            const int kb2 = (lane >> 4) << 1;
            const float* Pi = Op + (ti * 16 + m) * N_NODES + j * 16;
            const float* Pk = Op + (tk * 16 + m) * N_NODES + j * 16;
            v8f c = {};
            for (int k0 = 0; k0 < 16; k0 += 4) {
                v2f a = *(const v2f*)(Pi + k0 + kb2);
                v2f b = *(const v2f*)(Pk + k0 + kb2);
                c = wmma4(a, b, c);
            }
            const int col = lane & 15;
            const int rb8 = (lane >> 4) * 8;
            for (int r = 0; r < 8; ++r)
                Op[(ti * 16 + r + rb8) * N_NODES + tk * 16 + col] -= c[r];
        }
        __threadfence();
        __syncthreads();
    }
}

// ---------------------------------------------------------------------------
// Fused forward solve  L Y = F  (64 RHS), single workgroup of 32 waves.
// Diag block solved by 64 threads; solved 16x64 block staged in LDS; trailing
// GEMM updates (16x16x16 WMMA) distributed over the waves.
// ---------------------------------------------------------------------------
__global__ __launch_bounds__(1024)
void k_fsolve_fused(const float* __restrict__ Op, float* __restrict__ Y) {
    __shared__ float Lb[16][17];
    __shared__ float Yb[16][WIDTH];
    const int tid  = threadIdx.x;
    const int wave = tid >> 5;
    const int lane = tid & 31;

    for (int ib = 0; ib < NB; ++ib) {
        if (tid < 256)
            Lb[tid >> 4][tid & 15] =
                Op[(ib * 16 + (tid >> 4)) * N_NODES + ib * 16 + (tid & 15)];
        __syncthreads();

        if (tid < WIDTH) {
            float y[16];
            for (int r = 0; r < 16; ++r) {
                float s = Y[(ib * 16 + r) * WIDTH + tid];
                for (int c = 0; c < r; ++c) s -= Lb[r][c] * y[c];
                y[r] = s / Lb[r][r];
            }
            for (int r = 0; r < 16; ++r) {
                Yb[r][tid] = y[r];
                Y[(ib * 16 + r) * WIDTH + tid] = y[r];
            }
        }
        __syncthreads();

        const int nb = NB - 1 - ib;
        for (int t = wave; t < nb * 4; t += 32) {
            const int i  = ib + 1 + (t >> 2);
            const int n0 = (t & 3) * 16;
            const int m   = lane & 15;
            const int kb2 = (lane >> 4) << 1;
            const float* La = Op + (i * 16 + m) * N_NODES + ib * 16;
            v8f c = {};
            for (int k0 = 0; k0 < 16; k0 += 4) {
                v2f a = *(const v2f*)(La + k0 + kb2);
                v2f b;
                b.x = Yb[k0 + kb2    ][n0 + m];
                b.y = Yb[k0 + kb2 + 1][n0 + m];
                c = wmma4(a, b, c);
            }
            const int col = lane & 15;
            const int rb8 = (lane >> 4) * 8;
            for (int r = 0; r < 8; ++r)
                Y[(i * 16 + r + rb8) * WIDTH + n0 + col] -= c[r];
        }
        __threadfence();
        __syncthreads();
    }
}

// ---------------------------------------------------------------------------
// Fused backward solve  L^T X = Y, single workgroup of 32 waves.
// ---------------------------------------------------------------------------
__global__ __launch_bounds__(1024)
void k_bsolve_fused(const float* __restrict__ Op, float* __restrict__ Y) {
    __shared__ float Lb[16][17];
    __shared__ float Yb[16][WIDTH];
    const int tid  = threadIdx.x;
    const int wave = tid >> 5;
    const int lane = tid & 31;

    for (int ib = NB - 1; ib >= 0; --ib) {
        if (tid < 256)
            Lb[tid >> 4][tid & 15] =
                Op[(ib * 16 + (tid >> 4)) * N_NODES + ib * 16 + (tid & 15)];
        __syncthreads();

        if (tid < WIDTH) {
            float x[16];
            for (int r = 15; r >= 0; --r) {
                float s = Y[(ib * 16 + r) * WIDTH + tid];
                for (int c = r + 1; c < 16; ++c) s -= Lb[c][r] * x[c]; // U=L^T
                x[r] = s / Lb[r][r];
            }
            for (int r = 0; r < 16; ++r) {
                Yb[r][tid] = x[r];
                Y[(ib * 16 + r) * WIDTH + tid] = x[r];
            }
        }
        __syncthreads();

        for (int t = wave; t < ib * 4; t += 32) {
            const int kblk = t >> 2;
            const int n0   = (t & 3) * 16;
            const int m    = lane & 15;
            const int kb2  = (lane >> 4) << 1;
            v8f c = {};
            for (int k0 = 0; k0 < 16; k0 += 4) {
                v2f a;                       // A[m][k] = L[ib*16+k][kblk*16+m]
                a.x = Op[(ib * 16 + k0 + kb2    ) * N_NODES + kblk * 16 + m];
                a.y = Op[(ib * 16 + k0 + kb2 + 1) * N_NODES + kblk * 16 + m];
                v2f b;
                b.x = Yb[k0 + kb2    ][n0 + m];
                b.y = Yb[k0 + kb2 + 1][n0 + m];
                c = wmma4(a, b, c);
            }
            const int col = lane & 15;
            const int rb8 = (lane >> 4) * 8;
            for (int r = 0; r < 8; ++r)
                Y[(kblk * 16 + r + rb8) * WIDTH + n0 + col] -= c[r];
        }
        __threadfence();
        __syncthreads();
    }
}

// ---------------------------------------------------------------------------
// out[i][w] = leaky_relu( X[i][w] / max(dt[w], 1e-8) )
// ---------------------------------------------------------------------------
__global__ __launch_bounds__(256)
void k_final(const float* __restrict__ Y, const float* __restrict__ dt,
             float* __restrict__ out) {
    const int idx = blockIdx.x * 256 + threadIdx.x;
    if (idx < N_NODES * WIDTH) {
        const int w = idx & (WIDTH - 1);
        float v = Y[idx] / fmaxf(dt[w], 1e-8f);
        out[idx] = v > 0.0f ? v : LEAKY * v;
    }
}

extern "C" void kernel_launch(void* const* d_in, const int* in_sizes, int n_in,
                              void* d_out, int out_size, void* d_ws, size_t ws_size,
                              hipStream_t stream) {
    const float* node_fts = (const float*)d_in[0];
    const float* m_adv    = (const float*)d_in[3];
    const float* m_diff   = (const float*)d_in[4];
    const float* dt       = (const float*)d_in[7];
    const float* g_adv    = (const float*)d_in[8];
    const float* g_diff   = (const float*)d_in[9];
    const float* alpha    = (const float*)d_in[10];
    float* out = (float*)d_out;

    float* Op = (float*)d_ws;                          // 1024*1024 f32 = 4 MB
    float* Y  = Op + (size_t)N_NODES * N_NODES;        // 1024*64  f32 = 256 KB

    // 1) Op = ga*(Aa Aa^T)/n + gd*(Ad Ad^T)/n + (ga+gd+alpha)*I (lower tri)
    {
        dim3 g(N_NODES / 32, N_NODES / 32);
        k_build_op<<<g, 32, 0, stream>>>(m_adv, m_diff, g_adv, g_diff, alpha, Op);
    }

    // RHS working copy (do not mutate inputs)
    hipMemcpyAsync(Y, node_fts, (size_t)N_NODES * WIDTH * sizeof(float),
                   hipMemcpyDeviceToDevice, stream);

    // 2) Cholesky, 3) forward solve, 4) backward solve — fused persistent
    //    single-workgroup kernels (replaces ~450 dependent launches)
    k_chol_fused  <<<1, 1024, 0, stream>>>(Op);
    k_fsolve_fused<<<1, 1024, 0, stream>>>(Op, Y);
    k_bsolve_fused<<<1, 1024, 0, stream>>>(Op, Y);

    // 5) per-channel 1/dt scale + leaky ReLU
    k_final<<<(N_NODES * WIDTH + 255) / 256, 256, 0, stream>>>(Y, dt, out);
}